// BiMamba_1185410974355
// MI455X (gfx1250) — compile-verified
//
#include <hip/hip_runtime.h>
#include <math.h>

typedef __attribute__((ext_vector_type(2))) float v2f;
typedef __attribute__((ext_vector_type(8))) float v8f;

#define BSZ   2
#define NSITE 128
#define NCELL 8
#define DIM   32
#define D2    64
#define DI    128
#define DS    256
#define DR    4
#define LSEQ  1024
#define BL    2048           // BSZ * LSEQ rows
#define XPN   (DR + 2*DS)    // 516

#define ACT_NONE     0
#define ACT_RELU     1
#define ACT_SOFTPLUS 2

// CDNA5 async LDS path (guarded: falls back to register double-buffering)
#if defined(__HIP_DEVICE_COMPILE__) && defined(__gfx1250__) && \
    __has_builtin(__builtin_amdgcn_global_load_async_to_lds_b128) && \
    __has_builtin(__builtin_amdgcn_s_wait_asynccnt)
#define HAVE_ASYNC_LDS 1
// signature (from hipcc diagnostic): param0 = AS(1) int4*, param1 = AS(3) int4*,
// then two constant ints (offset, cpol)
typedef int v4i __attribute__((vector_size(16)));
typedef __attribute__((address_space(1))) v4i* gv4i_p;
typedef __attribute__((address_space(3))) v4i* lv4i_p;
#else
#define HAVE_ASYNC_LDS 0
#endif

__device__ __forceinline__ float sigf(float v) { return 1.0f / (1.0f + __expf(-v)); }

// ---------------- embedding concat + 2D positional encoding --------------
__global__ void embed_kernel(const float* __restrict__ x, const int* __restrict__ ycpg,
                             const int* __restrict__ cidx, const float* __restrict__ cellEB,
                             const float* __restrict__ CpGEB, float* __restrict__ hin)
{
    int row = blockIdx.x;             // b*1024 + s*8 + c
    int d   = threadIdx.x;            // 0..95
    int b = row >> 10;
    int r = row & 1023;
    int s = r >> 3;
    int c = r & 7;
    float v;
    if (d < 32) {
        int cls = ycpg[(b*NSITE + s)*NCELL + c];
        v = CpGEB[cls*DIM + d];
    } else if (d < 64) {
        int ci = cidx[b*NCELL + c];
        v = cellEB[ci*DIM + (d - 32)];
    } else {
        v = x[(b*NSITE + s)*DIM + (d - 64)];
    }
    // positionalencoding2d(96, 128, 8)[s, c, d]; dm = 48
    const float kfac = -logf(10000.0f) / 48.0f;
    float pe;
    if (d < 48) {
        int i = d >> 1;
        float div = expf((float)(2*i) * kfac);
        float arg = (float)c * div;
        pe = (d & 1) ? cosf(arg) : sinf(arg);
    } else {
        int j = d - 48;
        int i = j >> 1;
        float div = expf((float)(2*i) * kfac);
        float arg = (float)s * div;
        pe = (j & 1) ? cosf(arg) : sinf(arg);
    }
    hin[row*96 + d] = v + pe;
}

// ---------------- generic C = act(X @ W^T + bias) via f32 WMMA ------------
// X: M x K (row stride ldx), W: N x K (row stride ldw), C: M x N (row stride ldc)
// One wave32 per 16x16 output tile; K consumed 4 at a time by v_wmma_f32_16x16x4_f32.
__global__ void gemm_xwT(const float* __restrict__ X, int ldx,
                         const float* __restrict__ W, int ldw,
                         const float* __restrict__ bias,
                         float* __restrict__ C, int ldc,
                         int M, int N, int K, int act)
{
    int mt_cnt = M >> 4;
    int wid = blockIdx.x;
    int mt = wid % mt_cnt;
    int nt = wid / mt_cnt;
    int lane = threadIdx.x;
    int half = lane >> 4;            // 0: K={k0,k0+1}  1: K={k0+2,k0+3}
    int l    = lane & 15;

    int arow = mt*16 + l;            // A-matrix row  (M always multiple of 16)
    int ncol = nt*16 + l;            // B-matrix col
    int ncl  = (ncol < N) ? ncol : (N - 1);   // clamp loads; WMMA needs full EXEC

    const float* xp = X + (size_t)arow*ldx + 2*half;
    const float* wp = W + (size_t)ncl *ldw + 2*half;

    v8f acc = {0.f,0.f,0.f,0.f,0.f,0.f,0.f,0.f};
    for (int k0 = 0; k0 < K; k0 += 4) {
        if (k0 + 16 < K) {                       // global_prefetch_b8 next stripe
            __builtin_prefetch(xp + k0 + 16, 0, 3);
            __builtin_prefetch(wp + k0 + 16, 0, 3);
        }
        v2f a, b;
        a.x = xp[k0]; a.y = xp[k0 + 1];
        b.x = wp[k0]; b.y = wp[k0 + 1];
        acc = __builtin_amdgcn_wmma_f32_16x16x4_f32(
            /*neg_a=*/false, a, /*neg_b=*/false, b,
            /*c_mod=*/(short)0, acc, /*reuse_a=*/false, /*reuse_b=*/false);
    }

    // D layout: VGPR j -> M = 16*mt + 8*half + j, N = 16*nt + (lane&15)
    int om = mt*16 + half*8;
    int on = nt*16 + l;
    if (on < N) {
        float bv = bias ? bias[on] : 0.0f;
        #pragma unroll
        for (int j = 0; j < 8; ++j) {
            float v = acc[j] + bv;
            if (act == ACT_RELU)          v = fmaxf(v, 0.0f);
            else if (act == ACT_SOFTPLUS) v = (v > 20.0f) ? v : log1pf(__expf(v));
            C[(size_t)(om + j)*ldc + on] = v;
        }
    }
}

// ---------------- depthwise causal conv (K=4) + SiLU, both directions ----
// fwd: out[t] = sum_k w[k]*x[t-3+k] ; rev (flip-equivalent): sum_k w[k]*x[t+3-k]
// Writes xc2: rows [0,BL) = forward, rows [BL,2BL) = backward.
__global__ void conv_silu_kernel(const float* __restrict__ xz,
                                 const float* __restrict__ cw,
                                 const float* __restrict__ cb,
                                 float* __restrict__ xc2)
{
    int gid = blockIdx.x;            // dir*BL + b*L + t
    int rev = gid >= BL;
    int row = rev ? (gid - BL) : gid;
    int d   = threadIdx.x;           // 0..127
    int t    = row & (LSEQ - 1);
    int base = row - t;              // b*L
    float acc = cb[d];
    #pragma unroll
    for (int k = 0; k < 4; ++k) {
        int tt = rev ? (t + 3 - k) : (t - 3 + k);
        if (tt >= 0 && tt < LSEQ)
            acc += cw[d*4 + k] * xz[(size_t)(base + tt)*(2*DI) + d];
    }
    xc2[(size_t)gid*DI + d] = acc * sigf(acc);   // silu
}

// ---------------- selective scan: one wave32 per (dir, batch, channel) ---
// 256-wide SSM state in registers (8 f32/lane).  B/C rows double-buffered:
// async-to-LDS (ASYNCcnt) on gfx1250, register pipelining otherwise.
// Output fused with +x*D and the silu(z) gate.
__global__ void scan_kernel(const float* __restrict__ xc2,
                            const float* __restrict__ dt2,
                            const float* __restrict__ xd2,
                            const float* __restrict__ xz,
                            const float* __restrict__ A_log,
                            const float* __restrict__ Dp,
                            float* __restrict__ y2)
{
    int wid  = blockIdx.x;           // dir*256 + b*128 + d
    int dir  = wid >> 8;
    int b    = (wid >> 7) & 1;
    int d    = wid & 127;
    int lane = threadIdx.x;
    const int rev = dir;
    size_t rbase = (size_t)dir*BL + (size_t)b*LSEQ;   // row base in dir-split bufs

    float A[8], h[8];
    #pragma unroll
    for (int i = 0; i < 8; ++i) {
        A[i] = -__expf(A_log[(size_t)d*DS + lane*8 + i]);
        h[i] = 0.0f;
    }
    float Dv = Dp[d];

#if HAVE_ASYNC_LDS
    // sbuf[p][0][*] = B row, sbuf[p][1][*] = C row (contiguous 2KB per phase)
    __shared__ __align__(16) float sbuf[2][2][DS];
    // B|C occupy x_dbl cols [4,516): one base, four b128s at +0/+512/+1024/+1536B
    auto issue = [&](int p, int t) {
        const float* g = xd2 + (rbase + t)*XPN + DR + lane*4;
        gv4i_p gp = (gv4i_p)(unsigned long long)(uintptr_t)g;            // AS1, 64-bit
        lv4i_p lp = (lv4i_p)(unsigned)(uintptr_t)&sbuf[p][0][lane*4];    // AS3 = low 32b
        __builtin_amdgcn_global_load_async_to_lds_b128(gp, lp, 0,    0);
        __builtin_amdgcn_global_load_async_to_lds_b128(gp, lp, 512,  0);
        __builtin_amdgcn_global_load_async_to_lds_b128(gp, lp, 1024, 0);
        __builtin_amdgcn_global_load_async_to_lds_b128(gp, lp, 1536, 0);
    };
    issue(0, rev ? LSEQ - 1 : 0);
#else
    const float4* P0 = (const float4*)(xd2 + (rbase + (rev ? LSEQ-1 : 0))*XPN + DR + lane*8);
    float4 nB0 = P0[0], nB1 = P0[1], nC0 = P0[64], nC1 = P0[65];
#endif

    for (int step = 0; step < LSEQ; ++step) {
        int t = rev ? (LSEQ - 1 - step) : step;
        float4 B0, B1, C0, C1;
#if HAVE_ASYNC_LDS
        int p = step & 1;
        if (step + 1 < LSEQ) {
            issue(p ^ 1, rev ? (LSEQ - 2 - step) : (step + 1));
            __builtin_amdgcn_s_wait_asynccnt(4);   // oldest group (phase p) done
        } else {
            __builtin_amdgcn_s_wait_asynccnt(0);
        }
        const float4* Bp = (const float4*)&sbuf[p][0][lane*8];
        const float4* Cp = (const float4*)&sbuf[p][1][lane*8];
        B0 = Bp[0]; B1 = Bp[1]; C0 = Cp[0]; C1 = Cp[1];
#else
        B0 = nB0; B1 = nB1; C0 = nC0; C1 = nC1;
        if (step + 1 < LSEQ) {
            int tn = rev ? (LSEQ - 2 - step) : (step + 1);
            const float4* Q = (const float4*)(xd2 + (rbase + tn)*XPN + DR + lane*8);
            nB0 = Q[0]; nB1 = Q[1]; nC0 = Q[64]; nC1 = Q[65];
        }
#endif
        size_t rr = rbase + t;                       // dir-split row index
        float dtv = dt2[rr*DI + d];
        float xv  = xc2[rr*DI + d];
        float Bv[8] = {B0.x,B0.y,B0.z,B0.w,B1.x,B1.y,B1.z,B1.w};
        float Cv[8] = {C0.x,C0.y,C0.z,C0.w,C1.x,C1.y,C1.z,C1.w};
        float dx = dtv * xv;
        float part = 0.0f;
        #pragma unroll
        for (int i = 0; i < 8; ++i) {
            h[i] = __expf(dtv * A[i]) * h[i] + dx * Bv[i];
            part += h[i] * Cv[i];
        }
        #pragma unroll
        for (int off = 16; off >= 1; off >>= 1)
            part += __shfl_xor(part, off, 32);
        if (lane == 0) {
            float y = part + xv * Dv;
            float z = xz[(size_t)(b*LSEQ + t)*(2*DI) + DI + d];   // z shared by dirs
            y2[rr*DI + d] = y * (z * sigf(z));
        }
    }
}

// ---------------- residual + layernorm + (nsite,ncell) transpose ---------
// mode 0: src row (b, s*8+c)   -> dst row (b, c*128+s)
// mode 1: src row (b, c*128+s) -> dst row (b, s*8+c)
__global__ void ln_res_kernel(const float* __restrict__ h,
                              const float* __restrict__ of,
                              const float* __restrict__ ob,
                              const float* __restrict__ g,
                              const float* __restrict__ bta,
                              float* __restrict__ dst, int mode)
{
    int row  = blockIdx.x;
    int lane = threadIdx.x;          // 0..31, two elements each
    int d0 = lane, d1 = lane + 32;
    size_t i0 = (size_t)row*D2 + d0, i1 = (size_t)row*D2 + d1;
    float v0 = h[i0] + 0.5f*(of[i0] + ob[i0]);
    float v1 = h[i1] + 0.5f*(of[i1] + ob[i1]);
    float sum = v0 + v1;
    #pragma unroll
    for (int off = 16; off >= 1; off >>= 1) sum += __shfl_xor(sum, off, 32);
    float mu = sum * (1.0f/64.0f);
    float e0 = v0 - mu, e1 = v1 - mu;
    float vs = e0*e0 + e1*e1;
    #pragma unroll
    for (int off = 16; off >= 1; off >>= 1) vs += __shfl_xor(vs, off, 32);
    float rstd = rsqrtf(vs * (1.0f/64.0f) + 1e-5f);
    int b = row >> 10, r = row & 1023;
    int drow;
    if (mode == 0) { int s = r >> 3, c = r & 7;   drow = b*LSEQ + c*NSITE + s; }
    else           { int c = r >> 7, s = r & 127; drow = b*LSEQ + s*NCELL + c; }
    dst[(size_t)drow*D2 + d0] = e0*rstd*g[d0] + bta[d0];
    dst[(size_t)drow*D2 + d1] = e1*rstd*g[d1] + bta[d1];
}

static void launch_gemm(hipStream_t s, const float* X, int ldx, const float* W, int ldw,
                        const float* bias, float* C, int ldc, int M, int N, int K, int act)
{
    int grid = (M/16) * ((N + 15)/16);
    gemm_xwT<<<grid, 32, 0, s>>>(X, ldx, W, ldw, bias, C, ldc, M, N, K, act);
}

extern "C" void kernel_launch(void* const* d_in, const int* in_sizes, int n_in,
                              void* d_out, int out_size, void* d_ws, size_t ws_size,
                              hipStream_t stream)
{
    const float* x         = (const float*)d_in[0];
    const int*   ycpg      = (const int*)  d_in[1];
    const int*   cidx      = (const int*)  d_in[2];
    const float* cellEB    = (const float*)d_in[3];
    const float* CpGEB     = (const float*)d_in[4];
    const float* fcc_w     = (const float*)d_in[5];
    const float* fcc_b     = (const float*)d_in[6];
    const float* ln_g      = (const float*)d_in[7];
    const float* ln_b      = (const float*)d_in[8];
    const float* in_proj_w = (const float*)d_in[9];
    const float* conv_w    = (const float*)d_in[10];
    const float* conv_b    = (const float*)d_in[11];
    const float* x_proj_w  = (const float*)d_in[12];
    const float* dt_proj_w = (const float*)d_in[13];
    const float* dt_proj_b = (const float*)d_in[14];
    const float* A_log     = (const float*)d_in[15];
    const float* D_param   = (const float*)d_in[16];
    const float* out_proj_w= (const float*)d_in[17];

    float* ws   = (float*)d_ws;                  // ~18.8 MB total, L2-resident
    float* hin  = ws;                            // BL*96
    float* h1   = hin + (size_t)BL*96;           // BL*64
    float* h2   = h1  + (size_t)BL*D2;           // BL*64
    float* xzb  = h2  + (size_t)BL*D2;           // BL*256 (shared by both dirs)
    float* xc2  = xzb + (size_t)BL*2*DI;         // 2*BL*128  [fwd | bwd]
    float* xd2  = xc2 + (size_t)2*BL*DI;         // 2*BL*516
    float* dt2  = xd2 + (size_t)2*BL*XPN;        // 2*BL*128
    float* y2   = dt2 + (size_t)2*BL*DI;         // 2*BL*128
    float* o2   = y2  + (size_t)2*BL*DI;         // 2*BL*64   [out_f | out_b]

    // embeddings + posenc, then fcc: h = relu(hin @ fcc_w^T + fcc_b)
    embed_kernel<<<BL, 96, 0, stream>>>(x, ycpg, cidx, cellEB, CpGEB, hin);
    launch_gemm(stream, hin, 96, fcc_w, 96, fcc_b, h1, D2, BL, D2, 96, ACT_RELU);

    const float* hcur = h1;
    for (int layer = 0; layer < 2; ++layer) {
        float* dst = (layer == 0) ? h2 : (float*)d_out;
        // in_proj shared by both directions (pointwise => flip-invariant)
        launch_gemm(stream, hcur, D2, in_proj_w, D2, nullptr, xzb, 2*DI, BL, 2*DI, D2, ACT_NONE);
        // both directions fused from here: rows [0,BL)=fwd, [BL,2BL)=bwd
        conv_silu_kernel<<<2*BL, DI, 0, stream>>>(xzb, conv_w, conv_b, xc2);
        launch_gemm(stream, xc2, DI,  x_proj_w, DI, nullptr,    xd2, XPN, 2*BL, XPN, DI, ACT_NONE);
        launch_gemm(stream, xd2, XPN, dt_proj_w, DR, dt_proj_b, dt2, DI,  2*BL, DI,  DR, ACT_SOFTPLUS);
        scan_kernel<<<2*BSZ*DI, 32, 0, stream>>>(xc2, dt2, xd2, xzb, A_log, D_param, y2);
        launch_gemm(stream, y2, DI, out_proj_w, DI, nullptr, o2, D2, 2*BL, D2, DI, ACT_NONE);
        ln_res_kernel<<<BL, 32, 0, stream>>>(hcur, o2, o2 + (size_t)BL*D2, ln_g, ln_b, dst, layer);
        hcur = dst;
    }
}